// SmallAttention_6270652252548
// MI455X (gfx1250) — compile-verified
//
#include <hip/hip_runtime.h>

typedef __attribute__((ext_vector_type(2))) float v2f;
typedef __attribute__((ext_vector_type(8))) float v8f;

#define NVOX   4096     // 16^3
#define NHEADS 8
#define HDIM   32
#define CCH    256
#define BATCH  4

// D[M x Ncols] = A[M x K] * B[K x Ncols] (+ bias), exact fp32 via V_WMMA_F32_16X16X4_F32.
// One wave computes a 16x64 strip of D (4 x 16x16 tiles sharing one A-fragment),
// K consumed 4 at a time -> per k-step: 1 A-load + 8 B-loads feed 4 independent WMMAs.
// A-frag (16x4): lane L holds A[row0 + L%16][k + i + 2*(L/16)] in reg i (i=0,1)
// B-frag (4x16): lane L holds B[k + i + 2*(L/16)][col0 + L%16] in reg i
// C/D (16x16):   lane L reg i holds D[row0 + i + 8*(L/16)][col0 + L%16]
__global__ void __launch_bounds__(256)
wmma_gemm_f32(const float* A, const float* __restrict__ B,
              const float* __restrict__ bias, float* D,
              int M, int K, int Ncols, int lda) {
  const int lane = threadIdx.x & 31;
  const int wave = (blockIdx.x * blockDim.x + threadIdx.x) >> 5;
  const int ntn  = Ncols >> 6;                 // strips of 64 columns
  const int tm   = wave / ntn;
  const int tn   = wave - tm * ntn;
  if ((tm << 4) >= M) return;                  // wave-uniform: EXEC stays all-ones
  const int row0 = tm << 4, col0 = tn << 6;
  const int half = lane >> 4;                  // 0 or 1
  const int lid  = lane & 15;

  const float* arow = A + (size_t)(row0 + lid) * lda + (half << 1);
  const float* bcol = B + (size_t)(half << 1) * Ncols + col0 + lid;

  v8f acc0 = {0.f,0.f,0.f,0.f,0.f,0.f,0.f,0.f};
  v8f acc1 = acc0, acc2 = acc0, acc3 = acc0;

#pragma unroll 4
  for (int k = 0; k < K; k += 4) {
    v2f a = *(const v2f*)(arow + k);           // one 8B load, reused by 4 tiles
    const float* b0 = bcol + (size_t)k * Ncols;
    const float* b1 = b0 + Ncols;
    v2f bf0, bf1, bf2, bf3;
    bf0.x = b0[0];  bf0.y = b1[0];
    bf1.x = b0[16]; bf1.y = b1[16];
    bf2.x = b0[32]; bf2.y = b1[32];
    bf3.x = b0[48]; bf3.y = b1[48];
    acc0 = __builtin_amdgcn_wmma_f32_16x16x4_f32(false, a, false, bf0, (short)0, acc0, false, false);
    acc1 = __builtin_amdgcn_wmma_f32_16x16x4_f32(false, a, false, bf1, (short)0, acc1, false, false);
    acc2 = __builtin_amdgcn_wmma_f32_16x16x4_f32(false, a, false, bf2, (short)0, acc2, false, false);
    acc3 = __builtin_amdgcn_wmma_f32_16x16x4_f32(false, a, false, bf3, (short)0, acc3, false, false);
  }

  float* drow = D + (size_t)(row0 + (half << 3)) * Ncols + col0 + lid;
  const float bv0 = bias ? bias[col0 + lid]      : 0.0f;
  const float bv1 = bias ? bias[col0 + lid + 16] : 0.0f;
  const float bv2 = bias ? bias[col0 + lid + 32] : 0.0f;
  const float bv3 = bias ? bias[col0 + lid + 48] : 0.0f;
#pragma unroll
  for (int i = 0; i < 8; ++i) {
    float* dr = drow + (size_t)i * Ncols;
    dr[0]  = acc0[i] + bv0;
    dr[16] = acc1[i] + bv1;
    dr[32] = acc2[i] + bv2;
    dr[48] = acc3[i] + bv3;
  }
}

// One wave per (b, h, n); lane = channel within the 32-wide head (hd == wave32).
// k/v rows of neighbors are 32 contiguous floats -> coalesced 128B loads.
// Output written into the q-slot of qkv (q is dead after this kernel);
// each wave touches only its own (n, h*32..h*32+31) slice -> race-free.
__global__ void __launch_bounds__(256)
neigh_attn(const float* qkv, const int* __restrict__ neigh, float* out) {
  const int lane = threadIdx.x & 31;
  const int wave = (blockIdx.x * blockDim.x + threadIdx.x) >> 5;
  const int n = wave & (NVOX - 1);
  const int h = (wave >> 12) & (NHEADS - 1);
  const int b = wave >> 15;

  const float NEG_INF = -__builtin_inff();
  const size_t base = (size_t)(b * NVOX + n) * (3 * CCH);
  const float q = qkv[base + h * HDIM + lane] * 0.17677669529663687f; // hd^-0.5

  int   nb[27];
  float s[27];
#pragma unroll
  for (int j = 0; j < 27; ++j) {
    const int id = neigh[n * 27 + j];          // wave-uniform scalar
    nb[j] = id;
    float t = 0.f;
    if (id != NVOX) {
      const float* kr = qkv + (size_t)(b * NVOX + id) * (3 * CCH) + CCH + h * HDIM;
      t = q * kr[lane];
    }
#pragma unroll
    for (int off = 16; off; off >>= 1)         // butterfly: result in all 32 lanes
      t += __shfl_xor(t, off, 32);
    s[j] = (id == NVOX) ? NEG_INF : t;
  }

  float m = NEG_INF;
#pragma unroll
  for (int j = 0; j < 27; ++j) m = fmaxf(m, s[j]);
  float sum = 0.f;
#pragma unroll
  for (int j = 0; j < 27; ++j) { s[j] = __expf(s[j] - m); sum += s[j]; }
  const float inv = 1.0f / sum;

  float acc = 0.f;
#pragma unroll
  for (int j = 0; j < 27; ++j) {
    if (nb[j] != NVOX) {
      const float* vr = qkv + (size_t)(b * NVOX + nb[j]) * (3 * CCH) + 2 * CCH + h * HDIM;
      acc += s[j] * vr[lane];
    }
  }
  out[base + h * HDIM + lane] = acc * inv;     // overwrite dead q-slot
}

extern "C" void kernel_launch(void* const* d_in, const int* in_sizes, int n_in,
                              void* d_out, int out_size, void* d_ws, size_t ws_size,
                              hipStream_t stream) {
  (void)in_sizes; (void)n_in; (void)out_size; (void)ws_size;
  const float* x      = (const float*)d_in[0];   // (4, 4096, 256)
  const float* w_qkv  = (const float*)d_in[1];   // (256, 768)
  const float* w_proj = (const float*)d_in[2];   // (256, 256)
  const float* b_proj = (const float*)d_in[3];   // (256,)
  const int*   neigh  = (const int*)d_in[4];     // (4096, 27), pad == 4096
  float* out = (float*)d_out;                    // (4, 4096, 256)
  float* qkv = (float*)d_ws;                     // 16384 x 768 fp32 = 48 MB

  const int M = BATCH * NVOX;                    // 16384

  // 1) qkv = x @ w_qkv : (16384/16) * (768/64) = 12288 waves
  {
    int waves = (M / 16) * (3 * CCH / 64);
    wmma_gemm_f32<<<waves * 32 / 256, 256, 0, stream>>>(
        x, w_qkv, nullptr, qkv, M, CCH, 3 * CCH, CCH);
  }
  // 2) neighborhood attention: 4*8*4096 = 131072 waves
  {
    int waves = BATCH * NHEADS * NVOX;
    neigh_attn<<<waves * 32 / 256, 256, 0, stream>>>(qkv, neigh, qkv);
  }
  // 3) out = attn @ w_proj + b_proj : A strided inside qkv (lda = 768)
  {
    int waves = (M / 16) * (CCH / 64);
    wmma_gemm_f32<<<waves * 32 / 256, 256, 0, stream>>>(
        qkv, w_proj, b_proj, out, M, CCH, CCH, 3 * CCH);
  }
}